// TriHetGCN_65850438582348
// MI455X (gfx1250) — compile-verified
//
#include <hip/hip_runtime.h>

// ---------------------------------------------------------------------------
// 4-layer GCN (PyG GCNConv semantics) for gfx1250 / MI455X.
// GEMMs use V_WMMA_F32_16X16X4_F32 (full f32 accuracy; GEMM is ~20 GFLOP total
// and not the bottleneck). Message passing is the bandwidth-bound part: the
// 51.2 MB feature matrices are L2-resident (192 MB L2), gathers use coalesced
// float4 loads and f32 global atomics for the scatter-add.
// ---------------------------------------------------------------------------

#define NNODES 50000
#define NEDGES 400000
#define LN_EPS 1e-5f

typedef __attribute__((ext_vector_type(2))) float v2f;
typedef __attribute__((ext_vector_type(8))) float v8f;

// ------------------------------ small helpers ------------------------------

__global__ void zero_f32_kernel(float* __restrict__ p, int n) {
    int i = blockIdx.x * blockDim.x + threadIdx.x;
    if (i < n) p[i] = 0.0f;
}

__global__ void degree_kernel(const int* __restrict__ dst, float* __restrict__ deg, int e) {
    int i = blockIdx.x * blockDim.x + threadIdx.x;
    if (i < e) atomicAdd(&deg[dst[i]], 1.0f);
}

__global__ void dinv_kernel(float* __restrict__ d, int n) {
    int i = blockIdx.x * blockDim.x + threadIdx.x;
    if (i < n) d[i] = rsqrtf(d[i] + 1.0f);
}

// ------------------------------ WMMA GEMM ----------------------------------
// C[m0:m0+16, n0:n0+64] = A[m0:m0+16, 0:K] @ W[0:K, n0:n0+64]
// Block = 128 threads = 4 waves; wave w handles M-tile (blockIdx.x*4 + w).
// Epilogue writes hw = A@W and agg = hw * dinv[row]^2 (self-loop term).
//
// Fragment layouts (CDNA5 ISA 7.12.2, 32-bit, wave32):
//   A 16x4 : lanes 0-15 rows M=0..15 with K=0(v0),1(v1); lanes 16-31 K=2,3
//   B 4x16 : lanes 0-15 cols N=0..15 with K=0(v0),1(v1); lanes 16-31 K=2,3
//   C 16x16: vgpr r, lanes 0-15 -> M=r, lanes 16-31 -> M=8+r, N=lane&15
__global__ __launch_bounds__(128)
void gemm_wmma_f32_kernel(const float* __restrict__ A,
                          const float* __restrict__ W,
                          const float* __restrict__ dinv,
                          float* __restrict__ hw,
                          float* __restrict__ agg,
                          int K, int F) {
    const int tid  = threadIdx.x;
    const int wave = tid >> 5;
    const int lane = tid & 31;
    const int half = lane >> 4;       // 0: lanes 0-15, 1: lanes 16-31
    const int l15  = lane & 15;

    const int m0 = (blockIdx.x * 4 + wave) * 16;
    if (m0 >= NNODES) return;         // wave-uniform: EXEC stays all-ones
    const int n0 = blockIdx.y * 64;

    v8f c0 = {}, c1 = {}, c2 = {}, c3 = {};

    const float* arow = A + (size_t)(m0 + l15) * K + 2 * half;

    for (int kk = 0; kk < K; kk += 4) {
        // A fragment: 2 consecutive f32 (K and K+1 for this lane's half)
        v2f a = *(const v2f*)(arow + kk);

        // B fragments for the four 16-wide N sub-tiles
        const float* wb = W + (size_t)(kk + 2 * half) * F + n0 + l15;
        v2f b0, b1, b2, b3;
        b0.x = wb[0];  b0.y = wb[0  + F];
        b1.x = wb[16]; b1.y = wb[16 + F];
        b2.x = wb[32]; b2.y = wb[32 + F];
        b3.x = wb[48]; b3.y = wb[48 + F];

        c0 = __builtin_amdgcn_wmma_f32_16x16x4_f32(false, a, false, b0, (short)0, c0, false, false);
        c1 = __builtin_amdgcn_wmma_f32_16x16x4_f32(false, a, false, b1, (short)0, c1, false, false);
        c2 = __builtin_amdgcn_wmma_f32_16x16x4_f32(false, a, false, b2, (short)0, c2, false, false);
        c3 = __builtin_amdgcn_wmma_f32_16x16x4_f32(false, a, false, b3, (short)0, c3, false, false);
    }

#pragma unroll
    for (int r = 0; r < 8; ++r) {
        const int row = m0 + r + 8 * half;
        const float di = dinv[row];
        const float d2 = di * di;
        const size_t base = (size_t)row * F + n0 + l15;
        const float x0 = c0[r], x1 = c1[r], x2 = c2[r], x3 = c3[r];
        hw[base + 0]  = x0;  agg[base + 0]  = x0 * d2;
        hw[base + 16] = x1;  agg[base + 16] = x1 * d2;
        hw[base + 32] = x2;  agg[base + 32] = x2 * d2;
        hw[base + 48] = x3;  agg[base + 48] = x3 * d2;
    }
}

// --------------------------- edge scatter-add ------------------------------
// One thread per (edge, 4-feature chunk). Consecutive threads cover
// consecutive chunks of the same edge -> coalesced float4 gather of hw[src]
// and coalesced atomic stream into agg[dst]. Both matrices are L2-resident.
__global__ void scatter_edges_kernel(const int* __restrict__ src,
                                     const int* __restrict__ dst,
                                     const float* __restrict__ dinv,
                                     const float* __restrict__ hw,
                                     float* __restrict__ agg,
                                     int F4, int total) {
    int i = blockIdx.x * blockDim.x + threadIdx.x;
    if (i >= total) return;
    const int e = i / F4;
    const int c = i - e * F4;
    const int s = src[e];
    const int d = dst[e];
    const float w = dinv[s] * dinv[d];
    const float4 v = *((const float4*)hw + (size_t)s * F4 + c);
    float* ap = agg + ((size_t)d * F4 + c) * 4;
    atomicAdd(ap + 0, v.x * w);
    atomicAdd(ap + 1, v.y * w);
    atomicAdd(ap + 2, v.z * w);
    atomicAdd(ap + 3, v.w * w);
}

// ---------------------- bias + LayerNorm (+ ReLU) --------------------------
// One block per node, one thread per feature. In-place safe (block-local).
template <int F, bool RELU>
__global__ __launch_bounds__(F)
void ln_kernel(const float* __restrict__ in,
               const float* __restrict__ bias,
               const float* __restrict__ g,
               const float* __restrict__ be,
               float* __restrict__ out) {
    __shared__ float s[F];
    const int node = blockIdx.x;
    const int t = threadIdx.x;

    float v = in[(size_t)node * F + t] + bias[t];

    s[t] = v;
    __syncthreads();
#pragma unroll
    for (int st = F / 2; st > 0; st >>= 1) {
        if (t < st) s[t] += s[t + st];
        __syncthreads();
    }
    const float mean = s[0] * (1.0f / F);
    __syncthreads();

    const float dv = v - mean;
    s[t] = dv * dv;
    __syncthreads();
#pragma unroll
    for (int st = F / 2; st > 0; st >>= 1) {
        if (t < st) s[t] += s[t + st];
        __syncthreads();
    }
    const float var = s[0] * (1.0f / F);

    float y = dv * rsqrtf(var + LN_EPS) * g[t] + be[t];
    if (RELU) y = fmaxf(y, 0.0f);
    out[(size_t)node * F + t] = y;
}

// ------------------------------ launcher -----------------------------------

static inline void run_layer(const float* hin, const float* W, const float* b,
                             const float* gam, const float* bet,
                             const int* src, const int* dst, const float* dinv,
                             float* hwbuf, float* aggbuf, float* outbuf,
                             int K, int F, bool relu, hipStream_t stream) {
    dim3 gblk(128);
    dim3 ggrd((NNODES / 16 + 3) / 4, F / 64);
    gemm_wmma_f32_kernel<<<ggrd, gblk, 0, stream>>>(hin, W, dinv, hwbuf, aggbuf, K, F);

    const int F4 = F / 4;
    const int total = NEDGES * F4;
    scatter_edges_kernel<<<(total + 255) / 256, 256, 0, stream>>>(
        src, dst, dinv, hwbuf, aggbuf, F4, total);

    if (F == 256) {
        if (relu)
            ln_kernel<256, true><<<NNODES, 256, 0, stream>>>(aggbuf, b, gam, bet, outbuf);
        else
            ln_kernel<256, false><<<NNODES, 256, 0, stream>>>(aggbuf, b, gam, bet, outbuf);
    } else {
        if (relu)
            ln_kernel<128, true><<<NNODES, 128, 0, stream>>>(aggbuf, b, gam, bet, outbuf);
        else
            ln_kernel<128, false><<<NNODES, 128, 0, stream>>>(aggbuf, b, gam, bet, outbuf);
    }
}

extern "C" void kernel_launch(void* const* d_in, const int* in_sizes, int n_in,
                              void* d_out, int out_size, void* d_ws, size_t ws_size,
                              hipStream_t stream) {
    const float* x   = (const float*)d_in[0];
    const int*   ei  = (const int*)d_in[1];
    const int*   src = ei;
    const int*   dst = ei + NEDGES;
    const float* W1 = (const float*)d_in[2];  const float* b1 = (const float*)d_in[3];
    const float* W2 = (const float*)d_in[4];  const float* b2 = (const float*)d_in[5];
    const float* W3 = (const float*)d_in[6];  const float* b3 = (const float*)d_in[7];
    const float* W4 = (const float*)d_in[8];  const float* b4 = (const float*)d_in[9];
    const float* g1 = (const float*)d_in[10]; const float* be1 = (const float*)d_in[11];
    const float* g2 = (const float*)d_in[12]; const float* be2 = (const float*)d_in[13];
    const float* g3 = (const float*)d_in[14]; const float* be3 = (const float*)d_in[15];
    const float* g4 = (const float*)d_in[16]; const float* be4 = (const float*)d_in[17];
    float* out = (float*)d_out;

    // Workspace layout (floats):
    //   dinv : NNODES            (padded to 64)
    //   hw   : NNODES * 256
    //   bufA : NNODES * 256
    //   bufB : NNODES * 256
    float* ws = (float*)d_ws;
    const size_t npad = ((size_t)NNODES + 63) & ~(size_t)63;
    const size_t feat = (size_t)NNODES * 256;
    float* dinv = ws;
    float* hw   = ws + npad;
    float* bufA = hw + feat;
    float* bufB = bufA + feat;

    // Degree / normalization (recomputed every call: deterministic).
    zero_f32_kernel<<<(NNODES + 255) / 256, 256, 0, stream>>>(dinv, NNODES);
    degree_kernel<<<(NEDGES + 255) / 256, 256, 0, stream>>>(dst, dinv, NEDGES);
    dinv_kernel<<<(NNODES + 255) / 256, 256, 0, stream>>>(dinv, NNODES);

    // Layer 1: x[N,128] -> bufA[N,256]   (LN in-place on agg buffer)
    run_layer(x,    W1, b1, g1, be1, src, dst, dinv, hw, bufA, bufA, 128, 256, true,  stream);
    // Layer 2: bufA -> bufB
    run_layer(bufA, W2, b2, g2, be2, src, dst, dinv, hw, bufB, bufB, 256, 256, true,  stream);
    // Layer 3: bufB -> bufA
    run_layer(bufB, W3, b3, g3, be3, src, dst, dinv, hw, bufA, bufA, 256, 256, true,  stream);
    // Layer 4: bufA -> d_out[N,128] (agg in bufB, no ReLU)
    run_layer(bufA, W4, b4, g4, be4, src, dst, dinv, hw, bufB, out,  256, 128, false, stream);
}